// Aggregation_42322607734845
// MI455X (gfx1250) — compile-verified
//
#include <hip/hip_runtime.h>
#include <hip/hip_bf16.h>

// Aggregation / involution-style op:
//   out[b, g*16+wc, oh, ow] = sum_{idx} x[b, g*16+wc, oh+dy, ow+dx] * w[b, wc, idx, oh*64+ow]
// B=8, C=256, H=W=OH=OW=64, WC=16, G=16, K=3, PAD=1.
//
// Memory-bound (~86 MB traffic, ~0.15 GFLOP): no GEMM factorization exists
// (weighting is diagonal in channels), so no WMMA. We instead use the CDNA5
// async global->LDS data mover to stage the per-row weight slice (36 KB) and
// reuse it 4x per thread from LDS, with s_wait_asynccnt for completion.

#define AGG_B   8
#define AGG_C   256
#define AGG_HW  64
#define AGG_WC  16
#define AGG_K2  9
#define AGG_P   4096   // 64*64 pixels

typedef __attribute__((address_space(3))) float lds_f32;

__global__ __launch_bounds__(256) void aggregation_kernel(
    const float* __restrict__ x,     // [8,256,64,64]
    const float* __restrict__ wgt,   // [8,16,9,4096]
    float* __restrict__ out)         // [8,256,64,64]
{
    // weight slice for (b, oh): [wc*9+idx][ow]  -> 144*64 floats = 36 KB
    __shared__ float wLds[AGG_WC * AGG_K2 * AGG_HW];

    const int tid = threadIdx.x;          // 0..255
    const int bid = blockIdx.x;           // 0..511
    const int b   = bid >> 6;             // 0..7
    const int oh  = bid & 63;             // 0..63

    // ---------------- async stage weights into LDS ----------------
    // Flat copy of 9216 floats; each thread moves 9 x 16 bytes.
    {
        const float* wsrc = wgt + (size_t)b * (AGG_WC * AGG_K2 * AGG_P)
                                + (size_t)oh * AGG_HW;
#pragma unroll
        for (int pass = 0; pass < 9; ++pass) {
            const int f   = pass * 1024 + tid * 4;   // float index, 16B aligned
            const int row = f >> 6;                  // wc*9+idx  (0..143)
            const int owi = f & 63;
            const unsigned long long gaddr =
                (unsigned long long)(const void*)(wsrc + (size_t)row * AGG_P + owi);
            const unsigned ldsoff = (unsigned)(size_t)(lds_f32*)(&wLds[f]);
            asm volatile("global_load_async_to_lds_b128 %0, %1, off"
                         :: "v"(ldsoff), "v"(gaddr) : "memory");
        }
        asm volatile("s_wait_asynccnt 0" ::: "memory");
    }
    __syncthreads();

    // ---------------- compute ----------------
    const int ow = tid & 63;              // pixel column owned by this thread
    const int cq = tid >> 6;              // channel quarter (0..3)
    const int c0 = cq << 6;               // base channel for this quarter

    const bool row_up = (oh > 0);
    const bool row_dn = (oh < 63);
    const bool col_lf = (ow > 0);
    const bool col_rt = (ow < 63);

    const size_t base_b = (size_t)b * AGG_C * AGG_P;   // x/out batch base
    const int    p      = oh * AGG_HW + ow;            // pixel offset

    for (int wc = 0; wc < AGG_WC; ++wc) {
        // 9 per-pixel weights for this wc (conflict-free LDS reads: ow is
        // consecutive across lanes -> consecutive banks)
        float wv[9];
#pragma unroll
        for (int idx = 0; idx < AGG_K2; ++idx)
            wv[idx] = wLds[(wc * AGG_K2 + idx) * AGG_HW + ow];

        // reuse the weights for the 4 groups this quarter owns
#pragma unroll
        for (int gi = 0; gi < 4; ++gi) {
            const int c = c0 + gi * 16 + wc;           // c % 16 == wc
            const float* xp = x + base_b + (size_t)c * AGG_P + p;
            float acc = 0.0f;

            if (row_up) {                               // dy = -1, idx 0..2
                const float* r = xp - AGG_HW;
                if (col_lf) acc = fmaf(r[-1], wv[0], acc);
                acc = fmaf(r[0], wv[1], acc);
                if (col_rt) acc = fmaf(r[1], wv[2], acc);
            }
            {                                           // dy = 0, idx 3..5
                const float* r = xp;
                if (col_lf) acc = fmaf(r[-1], wv[3], acc);
                acc = fmaf(r[0], wv[4], acc);
                if (col_rt) acc = fmaf(r[1], wv[5], acc);
            }
            if (row_dn) {                               // dy = +1, idx 6..8
                const float* r = xp + AGG_HW;
                if (col_lf) acc = fmaf(r[-1], wv[6], acc);
                acc = fmaf(r[0], wv[7], acc);
                if (col_rt) acc = fmaf(r[1], wv[8], acc);
            }

            out[base_b + (size_t)c * AGG_P + p] = acc;
        }
    }
}

extern "C" void kernel_launch(void* const* d_in, const int* in_sizes, int n_in,
                              void* d_out, int out_size, void* d_ws, size_t ws_size,
                              hipStream_t stream) {
    const float* x   = (const float*)d_in[0];   // [8,256,64,64] fp32
    const float* wgt = (const float*)d_in[1];   // [8,16,9,4096] fp32
    float* out = (float*)d_out;                 // [8,256,64,64] fp32

    dim3 grid(AGG_B * AGG_HW);   // 512 blocks: one (b, oh) row each
    dim3 block(256);             // 8 wave32 waves
    aggregation_kernel<<<grid, block, 0, stream>>>(x, wgt, out);
}